// GraphConvolution_5798205850094
// MI455X (gfx1250) — compile-verified
//
#include <hip/hip_runtime.h>
#include <hip/hip_bf16.h>

typedef float v2f __attribute__((ext_vector_type(2)));
typedef float v8f __attribute__((ext_vector_type(8)));

#define DIM 256

// ---------------------------------------------------------------------------
// out[n][d] = bias[d]   (accumulator init; SpMM scatters on top of this)
// ---------------------------------------------------------------------------
__global__ void gc_bias_init(float* __restrict__ out,
                             const float* __restrict__ bias,
                             long long total) {
    long long i = (long long)blockIdx.x * blockDim.x + threadIdx.x;
    if (i < total) out[i] = bias[i & (DIM - 1)];
}

// ---------------------------------------------------------------------------
// h = x @ w  via V_WMMA_F32_16X16X4_F32 (exact fp32).
// Block = 256 threads = 8 waves. Block computes a 16 x 128 panel of h.
//   blockIdx.x : M tile (16 rows of x)          [12500 tiles, exact]
//   blockIdx.y : which 128-column half of D
//   blockIdx.z : adjacency (selects w0/h0 vs w1/h1)
// K is processed in 64-wide chunks staged in LDS (A: 16x64, B: 64x128).
// ---------------------------------------------------------------------------
__global__ void __launch_bounds__(256)
gc_gemm_wmma(const float* __restrict__ x,
             const float* __restrict__ w0,
             const float* __restrict__ w1,
             float* __restrict__ h0,
             float* __restrict__ h1,
             int n_rows) {
    // padded strides chosen for conflict-free WMMA operand reads
    __shared__ float la[16 * 68];    // A tile, stride 68
    __shared__ float lb[64 * 136];   // B panel, stride 136

    const int mtile = blockIdx.x;
    const int ncol0 = blockIdx.y * 128;
    const float* __restrict__ w = blockIdx.z ? w1 : w0;
    float* __restrict__ h       = blockIdx.z ? h1 : h0;

    const int t    = threadIdx.x;
    const int lane = t & 31;
    const int wv   = t >> 5;            // wave id 0..7 -> 16-col subtile
    const int m    = lane & 15;         // A row / B col / D col (per ISA layout)
    const int kh   = (lane >> 4) << 1;  // 0 for lanes 0-15, 2 for lanes 16-31

    v8f c = {0.f, 0.f, 0.f, 0.f, 0.f, 0.f, 0.f, 0.f};

    for (int kc = 0; kc < DIM; kc += 64) {
        // stage A: 16 rows x 64 K-values (1024 floats, 1 float4 per thread)
        {
            const int row = t >> 4;            // 0..15
            const int c4  = (t & 15) << 2;     // 0..60
            const float4 v = *(const float4*)(x + (size_t)(mtile * 16 + row) * DIM + kc + c4);
            *(float4*)(&la[row * 68 + c4]) = v;
        }
        // stage B: 64 K-rows x 128 cols (8192 floats, 8 float4 per thread)
        #pragma unroll
        for (int i = 0; i < 8; ++i) {
            const int idx = t + i * 256;       // 0..2047 float4 slots
            const int kr  = idx >> 5;          // 0..63
            const int c4  = (idx & 31) << 2;   // 0..124
            const float4 v = *(const float4*)(w + (size_t)(kc + kr) * DIM + ncol0 + c4);
            *(float4*)(&lb[kr * 136 + c4]) = v;
        }
        __syncthreads();

        #pragma unroll
        for (int ks = 0; ks < 64; ks += 4) {
            // A operand: lane holds A[m][ks+kh], A[m][ks+kh+1] (contiguous -> ds_load_b64)
            v2f a = *(const v2f*)(&la[m * 68 + ks + kh]);
            // B operand: lane holds B[ks+kh][n], B[ks+kh+1][n], n = wv*16 + m
            const int n = wv * 16 + m;
            v2f b;
            b.x = lb[(ks + kh) * 136 + n];
            b.y = lb[(ks + kh + 1) * 136 + n];
            c = __builtin_amdgcn_wmma_f32_16x16x4_f32(
                    /*neg_a=*/false, a, /*neg_b=*/false, b,
                    /*c_mod=*/(short)0, c, /*reuse_a=*/false, /*reuse_b=*/false);
        }
        __syncthreads();
    }

    // D layout: VGPR r, lanes 0-15 -> row r, lanes 16-31 -> row r+8; col = lane&15
    const int ncol = ncol0 + wv * 16 + m;
    const int mrow = mtile * 16 + ((lane >> 4) << 3);
    #pragma unroll
    for (int r = 0; r < 8; ++r) {
        h[(size_t)(mrow + r) * DIM + ncol] = c[r];
    }
    (void)n_rows;
}

// ---------------------------------------------------------------------------
// out[rows[e]][:] += vals[e] * h[cols[e]][:]
// 64 threads per edge (edge index is wave-uniform: a wave32 covers one half
// of an edge's 256 floats). Edge metadata is read through the scalar path
// (readfirstlane -> s_load) and the payload uses float4 gathers plus
// hardware f32 atomics (unsafeAtomicAdd -> global_atomic_add_f32 in L2).
// ---------------------------------------------------------------------------
__global__ void gc_spmm_scatter(const int* __restrict__ rows,
                                const int* __restrict__ cols,
                                const float* __restrict__ vals,
                                const float* __restrict__ h,
                                float* __restrict__ out,
                                int nnz) {
    const unsigned long long idx =
        (unsigned long long)blockIdx.x * blockDim.x + threadIdx.x;
    const int e = (int)(idx >> 6);
    if (e >= nnz) return;
    // e is uniform across the wave -> force scalar loads for edge metadata
    const int e_s = __builtin_amdgcn_readfirstlane(e);
    const int r   = rows[e_s];
    const int cl  = cols[e_s];
    const float v = vals[e_s];

    const int c = (int)(idx & 63);
    const float4 msg = *(const float4*)(h + (size_t)cl * DIM + c * 4);
    float* dst = out + (size_t)r * DIM + c * 4;
    unsafeAtomicAdd(dst + 0, v * msg.x);
    unsafeAtomicAdd(dst + 1, v * msg.y);
    unsafeAtomicAdd(dst + 2, v * msg.z);
    unsafeAtomicAdd(dst + 3, v * msg.w);
}

// ---------------------------------------------------------------------------
// out = tanh(out)
// ---------------------------------------------------------------------------
__global__ void gc_tanh(float* __restrict__ out, long long total) {
    long long i = (long long)blockIdx.x * blockDim.x + threadIdx.x;
    if (i < total) out[i] = tanhf(out[i]);
}

extern "C" void kernel_launch(void* const* d_in, const int* in_sizes, int n_in,
                              void* d_out, int out_size, void* d_ws, size_t ws_size,
                              hipStream_t stream) {
    const float* x     = (const float*)d_in[0];
    const float* w0    = (const float*)d_in[1];
    const float* w1    = (const float*)d_in[2];
    const float* bias  = (const float*)d_in[3];
    const float* vals0 = (const float*)d_in[4];
    const float* vals1 = (const float*)d_in[5];
    const int*   rows0 = (const int*)d_in[6];
    const int*   cols0 = (const int*)d_in[7];
    const int*   rows1 = (const int*)d_in[8];
    const int*   cols1 = (const int*)d_in[9];
    float* out = (float*)d_out;

    const int n_rows = in_sizes[0] / DIM;    // 200000
    const int nnz    = in_sizes[4];          // 6400000
    const long long total = (long long)n_rows * DIM;

    float* h0 = (float*)d_ws;
    float* h1 = h0 + (size_t)n_rows * DIM;
    (void)ws_size;

    // 1) out = bias (broadcast)
    {
        const int blocks = (int)((total + 255) / 256);
        gc_bias_init<<<blocks, 256, 0, stream>>>(out, bias, total);
    }

    // 2) h0 = x@w0, h1 = x@w1 (WMMA f32)
    {
        dim3 grid(n_rows / 16, 2, 2);        // 12500 x (2 col-halves) x (2 adjacencies)
        gc_gemm_wmma<<<grid, 256, 0, stream>>>(x, w0, w1, h0, h1, n_rows);
    }

    // 3) scatter-add both adjacencies
    {
        const unsigned long long threads = (unsigned long long)nnz * 64ull;
        const int blocks = (int)((threads + 255ull) / 256ull);
        gc_spmm_scatter<<<blocks, 256, 0, stream>>>(rows0, cols0, vals0, h0, out, nnz);
        gc_spmm_scatter<<<blocks, 256, 0, stream>>>(rows1, cols1, vals1, h1, out, nnz);
    }

    // 4) out = tanh(out)
    {
        const int blocks = (int)((total + 255) / 256);
        gc_tanh<<<blocks, 256, 0, stream>>>(out, total);
    }
}